// CBOWNegativeSampling_73014444032055
// MI455X (gfx1250) — compile-verified
//
#include <hip/hip_runtime.h>

// CDNA5 / gfx1250: wave32, WMMA 16x16x4 f32
typedef float v2f __attribute__((ext_vector_type(2)));
typedef float v8f __attribute__((ext_vector_type(8)));

#define VOCAB 100000
#define EMB   300
#define BATCH 16384
#define CTX   10
#define NEGK  20
#define ROWS  (CTX + NEGK)   // 30 gathered o_emb rows per batch element
#define WAVES_PER_BLOCK 8    // 256 threads

// Fast log(sigmoid(x)) = min(x,0) - log(1 + exp(-|x|))
// Uses hardware v_exp_f32 / v_log_f32 (single-instruction transcendentals).
__device__ __forceinline__ float log_sigmoid_fast(float x) {
    return fminf(x, 0.0f) - __logf(1.0f + __expf(-fabsf(x)));
}

__global__ void cbow_zero_out(float* out) {
    if (threadIdx.x == 0) out[0] = 0.0f;
}

__global__ __launch_bounds__(WAVES_PER_BLOCK * 32)
void CBOWNegativeSampling_kernel(const float* __restrict__ i_emb,
                                 const float* __restrict__ o_emb,
                                 const int*   __restrict__ context,
                                 const int*   __restrict__ target,
                                 const int*   __restrict__ neg_samples,
                                 float*       __restrict__ out) {
    const int lane = threadIdx.x & 31;
    const int wave = threadIdx.x >> 5;
    const int b    = blockIdx.x * WAVES_PER_BLOCK + wave;   // one wave per batch element

    const int half = lane >> 4;          // 0: K{0,1}, 1: K{2,3} of each WMMA K-step
    const int mlan = lane & 15;
    const int koff = half * 2;

    __shared__ float block_sum;
    if (threadIdx.x == 0) block_sum = 0.0f;

    // ---- gather row pointers ----
    const float* tgt = i_emb + (size_t)target[b] * EMB;

    // Tile 0 covers rows 0..15 (10 ctx + first 6 neg); tile 1 rows 16..29 (+2 pad)
    const int m0 = mlan;
    const int m1 = 16 + mlan;
    const int r0 = (m0 < CTX)  ? context[b * CTX + m0]
                               : neg_samples[b * NEGK + (m0 - CTX)];
    const int r1 = (m1 < ROWS) ? neg_samples[b * NEGK + (m1 - CTX)] : 0; // pad -> row 0
    const float* a0p = o_emb + (size_t)r0 * EMB;
    const float* a1p = o_emb + (size_t)r1 * EMB;

    // gfx1250 global_prefetch_b8 on the gathered rows
    __builtin_prefetch(tgt, 0, 3);
    __builtin_prefetch(a0p, 0, 3);
    __builtin_prefetch(a1p, 0, 3);

    __syncthreads();   // block_sum initialized before any wave adds to it

    // ---- WMMA K-loop: D = A(16x4,f32) x B(4x16,f32) + C, 75 steps over D=300 ----
    // B columns are all identical (broadcast target chunk), so every column of D
    // holds the same 16 dot products.
    v8f c0 = {};
    v8f c1 = {};
#pragma unroll 5
    for (int kb = 0; kb < EMB; kb += 4) {
        const v2f bf = *(const v2f*)(tgt + kb + koff);  // same addr per half-wave: broadcast
        const v2f a0 = *(const v2f*)(a0p + kb + koff);
        const v2f a1 = *(const v2f*)(a1p + kb + koff);
        // 8 args: (neg_a, A, neg_b, B, c_mod, C, reuse_a, reuse_b)
        c0 = __builtin_amdgcn_wmma_f32_16x16x4_f32(false, a0, false, bf, (short)0, c0, false, false);
        c1 = __builtin_amdgcn_wmma_f32_16x16x4_f32(false, a1, false, bf, (short)0, c1, false, false);
    }

    // ---- epilogue ----
    // C/D f32 16x16 layout: lanes 0-15 hold M = v (VGPR v), lanes 16-31 hold M = 8+v.
    // So half 0 owns rows {0..7, 16..23}; half 1 owns rows {8..15, 24..29}.
    const int base0 = half ? 8 : 0;
    float acc = 0.0f;
#pragma unroll
    for (int v = 0; v < 8; ++v) {
        const int m = base0 + v;              // rows 0..15
        const float d = c0[v];
        acc += (m < CTX) ? log_sigmoid_fast(d) * (1.0f / CTX)  // positive: mean over ctx
                         : log_sigmoid_fast(-d);               // negative: sum logsig(-logit)
    }
#pragma unroll
    for (int v = 0; v < 8; ++v) {
        const int m = 16 + base0 + v;         // rows 16..31 (30,31 are padding)
        if (m < ROWS) acc += log_sigmoid_fast(-c1[v]);         // all >= 16 are negatives
    }

    // acc is uniform within each 16-lane half; combine halves -> per-batch loss term.
    const float total = acc + __shfl_xor(acc, 16, 32);

    // Hierarchical reduction: per-wave leader -> LDS (ds_add_f32) -> one global atomic.
    if (lane == 0) {
        atomicAdd(&block_sum, -total * (1.0f / (float)BATCH));
    }
    __syncthreads();
    if (threadIdx.x == 0) {
        atomicAdd(out, block_sum);
    }
}

extern "C" void kernel_launch(void* const* d_in, const int* in_sizes, int n_in,
                              void* d_out, int out_size, void* d_ws, size_t ws_size,
                              hipStream_t stream) {
    const float* i_emb       = (const float*)d_in[0];
    const float* o_emb       = (const float*)d_in[1];
    const int*   context     = (const int*)d_in[2];
    const int*   target      = (const int*)d_in[3];
    const int*   neg_samples = (const int*)d_in[4];
    float*       out         = (float*)d_out;

    cbow_zero_out<<<1, 32, 0, stream>>>(out);

    const int blocks = BATCH / WAVES_PER_BLOCK;   // 2048
    CBOWNegativeSampling_kernel<<<blocks, WAVES_PER_BLOCK * 32, 0, stream>>>(
        i_emb, o_emb, context, target, neg_samples, out);
}